// Attention_21680994910931
// MI455X (gfx1250) — compile-verified
//
#include <hip/hip_runtime.h>
#include <math.h>

// ---------------------------------------------------------------------------
// Problem constants (from the reference)
// ---------------------------------------------------------------------------
#define HEADS 8
#define DIMM  512
#define DH    64
#define BB_   2
#define NN_   512
#define SS_   1536            // 3*N (concat along sequence)
#define M_ROWS (BB_*SS_)      // 3072
#define QKV_N  (3*HEADS*DH)   // 1536
#define HD     (HEADS*DH)     // 512
#define KTILES (SS_/16)       // 96 key tiles
#define NWAVE  4              // waves per attention block
#define TPW    (KTILES/NWAVE) // 24 key tiles per wave
#define KSTR   68             // padded K-tile LDS row stride (DWORDs)

typedef float v2f __attribute__((ext_vector_type(2)));
typedef float v8f __attribute__((ext_vector_type(8)));
typedef unsigned int u32x4 __attribute__((ext_vector_type(4)));
typedef int i32x4 __attribute__((ext_vector_type(4)));
typedef int i32x8 __attribute__((ext_vector_type(8)));

// D = A(16x4 f32) * B(4x16 f32) + C(16x16 f32)  -> v_wmma_f32_16x16x4_f32
__device__ __forceinline__ v8f wmma4(v2f a, v2f b, v8f c) {
  return __builtin_amdgcn_wmma_f32_16x16x4_f32(
      false, a, false, b, (short)0, c, false, false);
}

// TDM group-1 word0: data_size=4B (code 2). K tiles add LDS padding:
// pad_enable, pad_interval code 5 (=64 DWORDs), pad_amount code 3 (=4 DWORDs)
// -> 68-DWORD padded rows in LDS (bank-conflict-free column reads).
#define TDM_CFG_K ((2<<16) | (1<<20) | (5<<22) | (3<<25))
#define TDM_CFG_V (2<<16)

// One 16x64-f32 tile: global (row stride 64) -> LDS via Tensor Data Mover.
// This toolchain's builtin is the 6-arg form:
//   (u32x4 g0, i32x8 g1, i32x4 g2, i32x4 g3, i32x8 g4, i32 cpol)
__device__ __forceinline__ void tdm_load_tile(unsigned lds_addr,
                                              const float* gptr, int cfg) {
  unsigned long long ga = (unsigned long long)(uintptr_t)gptr;
  u32x4 g0;
  g0[0] = 1u;                                   // count=1, user mode
  g0[1] = lds_addr;                             // LDS byte address
  g0[2] = (unsigned)(ga & 0xFFFFFFFFu);         // global_addr[31:0]
  g0[3] = (unsigned)((ga >> 32) & 0x1FFFFFFu)   // global_addr[56:32]
        | (2u << 30);                           // type = 2 ("image")
  i32x8 g1;
  g1[0] = cfg;                                  // mask=0, data_size, pad cfg
  g1[1] = (int)(64u << 16);                     // tensor_dim0 = 64
  g1[2] = (int)(16u << 16);                     // tensor_dim1 = 16
  g1[3] = (int)(64u << 16);                     // tile_dim0 = 64
  g1[4] = 16;                                   // tile_dim1 = 16
  g1[5] = 64;                                   // tensor_dim0_stride = 64
  g1[6] = 0;
  g1[7] = 0;
  i32x4 z4 = {0, 0, 0, 0};
  i32x8 z8 = {0, 0, 0, 0, 0, 0, 0, 0};
  __builtin_amdgcn_tensor_load_to_lds(g0, g1, z4, z4, z8, 0);
}

// ---------------------------------------------------------------------------
// Kernel 1: qkv = concat(q,k,v) @ W_qkv, scattered into head-major Q/K/V.
// ---------------------------------------------------------------------------
__global__ __launch_bounds__(32) void qkv_gemm_kernel(
    const float* __restrict__ q, const float* __restrict__ k,
    const float* __restrict__ v, const float* __restrict__ W,
    float* __restrict__ Qh, float* __restrict__ Kh, float* __restrict__ Vh)
{
  const int lane = threadIdx.x;
  const int n16  = lane & 15;
  const int hi   = lane >> 4;
  const int tm   = blockIdx.x;     // 0..191
  const int tn   = blockIdx.y;     // 0..95

  const int m  = tm * 16 + n16;    // A-operand row (m = lane%16)
  const int b0 = m / SS_;
  const int r  = m % SS_;
  const float* arow;
  if (r < NN_)          arow = q + ((size_t)b0 * NN_ + r) * DIMM;
  else if (r < 2*NN_)   arow = k + ((size_t)b0 * NN_ + (r - NN_)) * DIMM;
  else                  arow = v + ((size_t)b0 * NN_ + (r - 2*NN_)) * DIMM;

  const int c = tn * 16 + n16;     // B-operand column (n = lane%16)
  const float* bcol = W + c;

  v8f acc = {0,0,0,0,0,0,0,0};
  #pragma unroll 8
  for (int kk = 0; kk < DIMM; kk += 4) {
    v2f a = *(const v2f*)(arow + kk + 2*hi);
    v2f b;
    b.x = bcol[(size_t)(kk + 2*hi)     * QKV_N];
    b.y = bcol[(size_t)(kk + 2*hi + 1) * QKV_N];
    acc = wmma4(a, b, acc);
  }

  // Scatter epilogue (tiles never straddle the batch boundary: 1536 % 16 == 0)
  const int chunk = c / HD;
  const int h     = (c % HD) / DH;
  const int d     = c % DH;
  float* dst = (chunk == 0) ? Qh : (chunk == 1) ? Kh : Vh;
  const int b2 = (tm * 16) / SS_;
  const int rb = (tm * 16) % SS_ + 8*hi;
  float* drow = dst + (((size_t)h * BB_ + b2) * SS_ + rb) * DH + d;
  #pragma unroll
  for (int vv = 0; vv < 8; ++vv) drow[(size_t)vv * DH] = acc[vv];
}

// ---------------------------------------------------------------------------
// Kernel 2: per-row statistics (mean, unbiased var, L2 norm) for Q and K.
// ---------------------------------------------------------------------------
__global__ void stats_kernel(
    const float* __restrict__ Qh, const float* __restrict__ Kh,
    float* __restrict__ qmean, float* __restrict__ qvar, float* __restrict__ qnorm,
    float* __restrict__ kmean, float* __restrict__ kvar, float* __restrict__ knorm)
{
  const int NROWS = HEADS * BB_ * SS_;       // 24576
  int t = blockIdx.x * blockDim.x + threadIdx.x;
  if (t >= 2 * NROWS) return;
  int which = t / NROWS;
  int idx   = t % NROWS;
  const float* row = (which ? Kh : Qh) + (size_t)idx * DH;
  float s = 0.f, s2 = 0.f;
  #pragma unroll
  for (int i = 0; i < DH; i += 4) {
    float4 x = *(const float4*)(row + i);
    s  += x.x + x.y + x.z + x.w;
    s2 += x.x*x.x + x.y*x.y + x.z*x.z + x.w*x.w;
  }
  float mean = s * (1.f / DH);
  float var  = (s2 - s * s * (1.f / DH)) * (1.f / (DH - 1));  // ddof=1
  float nrm  = sqrtf(s2);
  if (which) { kmean[idx] = mean; kvar[idx] = var; knorm[idx] = nrm; }
  else       { qmean[idx] = mean; qvar[idx] = var; qnorm[idx] = nrm; }
}

// ---------------------------------------------------------------------------
// Kernel 3: flash attention. 4 waves / block, each wave owns 24 key tiles
// (split-K online softmax), K/V tiles double-buffered into LDS by the TDM,
// partials (m,l,U) combined through LDS after a real workgroup barrier.
// ---------------------------------------------------------------------------
__global__ __launch_bounds__(NWAVE * 32) void flash_attn_kernel(
    const float* __restrict__ Qh, const float* __restrict__ Kh,
    const float* __restrict__ Vh,
    const float* __restrict__ qmean, const float* __restrict__ qvar,
    const float* __restrict__ qnorm,
    const float* __restrict__ kmean, const float* __restrict__ kvar,
    const float* __restrict__ knorm,
    const float* __restrict__ cw_raw, const float* __restrict__ vw_raw,
    float* __restrict__ Obuf)
{
  __shared__ float KbufS[NWAVE][2][16 * KSTR];  // TDM-padded K tiles
  __shared__ float VbufS[NWAVE][2][16 * 64];    // V tiles
  __shared__ float Ptile[NWAVE][16 * 18];       // P: C-layout -> A-layout
  __shared__ float Ubuf [NWAVE][16 * 64];       // per-wave unnormalized O
  __shared__ float Mbuf [NWAVE][16];            // per-wave row max
  __shared__ float Lbuf [NWAVE][16];            // per-wave row sum

  const int lane = threadIdx.x & 31;
  const int w    = __builtin_amdgcn_readfirstlane(threadIdx.x >> 5);
  const int n16  = lane & 15;
  const int hi   = lane >> 4;
  const int qt   = blockIdx.x;               // query tile 0..95
  const int hb   = blockIdx.y;               // 0..15
  const int h    = hb >> 1, b0 = hb & 1;
  const size_t rowoff = ((size_t)h * BB_ + b0) * SS_;

  const float cov_w = 1.f / (1.f + __expf(-cw_raw[0]));
  const float var_w = 1.f / (1.f + __expf(-vw_raw[0]));
  const float cos_w = 1.f - cov_w - var_w;
  const float cw64  = cov_w * (1.f / DH);
  const float vw64  = var_w * (1.f / DH);

  // Per-lane query-row stats for rows qt*16 + vv + 8*hi
  float qn_[8], qm_[8], qv_[8];
  #pragma unroll
  for (int vv = 0; vv < 8; ++vv) {
    size_t i = rowoff + qt * 16 + vv + 8*hi;
    qn_[vv] = qnorm[i]; qm_[vv] = qmean[i]; qv_[vv] = qvar[i];
  }

  // Preload the Q tile into registers (A-layout), reused for all 24 key tiles
  const float* Qrow = Qh + (rowoff + qt * 16 + n16) * DH;
  v2f qa[16];
  #pragma unroll
  for (int kk = 0; kk < DH; kk += 4)
    qa[kk >> 2] = *(const v2f*)(Qrow + kk + 2*hi);

  const unsigned kb0 = (unsigned)(uintptr_t)&KbufS[w][0][0];
  const unsigned kb1 = (unsigned)(uintptr_t)&KbufS[w][1][0];
  const unsigned vb0 = (unsigned)(uintptr_t)&VbufS[w][0][0];
  const unsigned vb1 = (unsigned)(uintptr_t)&VbufS[w][1][0];

  v8f acc0 = {0,0,0,0,0,0,0,0}, acc1 = acc0, acc2 = acc0, acc3 = acc0;
  float mrow[8], lrow[8];
  #pragma unroll
  for (int vv = 0; vv < 8; ++vv) { mrow[vv] = -__builtin_inff(); lrow[vv] = 0.f; }

  const int kt0 = w * TPW;
  // Prologue: TDM-load first K/V tile for this wave
  tdm_load_tile(kb0, Kh + (rowoff + (size_t)kt0 * 16) * DH, TDM_CFG_K);
  tdm_load_tile(vb0, Vh + (rowoff + (size_t)kt0 * 16) * DH, TDM_CFG_V);

  for (int it = 0; it < TPW; ++it) {
    const int kt  = kt0 + it;
    const int cur = it & 1;
    if (it + 1 < TPW) {   // issue next tile's DMA, then wait for current pair
      tdm_load_tile(cur ? kb0 : kb1,
                    Kh + (rowoff + (size_t)(kt + 1) * 16) * DH, TDM_CFG_K);
      tdm_load_tile(cur ? vb0 : vb1,
                    Vh + (rowoff + (size_t)(kt + 1) * 16) * DH, TDM_CFG_V);
      __builtin_amdgcn_s_wait_tensorcnt(2);
    } else {
      __builtin_amdgcn_s_wait_tensorcnt(0);
    }
    asm volatile("" ::: "memory");   // keep LDS reads behind the tensorcnt wait

    const float* Kl = &KbufS[w][cur][0];
    const float* Vl = &VbufS[w][cur][0];

    // S = Q_tile @ K_tile^T : two independent 8-deep WMMA chains (K=64)
    v8f sa = {0,0,0,0,0,0,0,0}, sb = sa;
    #pragma unroll
    for (int kk = 0; kk < 32; kk += 4)
      sa = wmma4(qa[kk >> 2], *(const v2f*)(Kl + n16 * KSTR + kk + 2*hi), sa);
    #pragma unroll
    for (int kk = 32; kk < 64; kk += 4)
      sb = wmma4(qa[kk >> 2], *(const v2f*)(Kl + n16 * KSTR + kk + 2*hi), sb);
    v8f s = sa + sb;

    // Key-column stats for this lane
    size_t j = rowoff + kt * 16 + n16;
    const float knj = knorm[j], kmj = kmean[j], kvj = kvar[j];

    float scl[8];
    #pragma unroll
    for (int vv = 0; vv < 8; ++vv) {
      float d = s[vv] * (cos_w / (qn_[vv] * knj + 1e-8f) + cw64)
              + vw64 * qv_[vv] * kvj
              - cov_w * qm_[vv] * kmj;
      float t = d;                       // row max over the 16-lane half
      t = fmaxf(t, __shfl_xor(t, 1, 32));
      t = fmaxf(t, __shfl_xor(t, 2, 32));
      t = fmaxf(t, __shfl_xor(t, 4, 32));
      t = fmaxf(t, __shfl_xor(t, 8, 32));
      float mnew = fmaxf(mrow[vv], t);
      float p = __expf(d - mnew);
      float r = p;
      r += __shfl_xor(r, 1, 32);
      r += __shfl_xor(r, 2, 32);
      r += __shfl_xor(r, 4, 32);
      r += __shfl_xor(r, 8, 32);
      float sc = __expf(mrow[vv] - mnew);
      lrow[vv] = lrow[vv] * sc + r;
      mrow[vv] = mnew;
      scl[vv]  = sc;
      Ptile[w][(vv + 8*hi) * 18 + n16] = p;   // C-layout -> LDS
    }
    #pragma unroll
    for (int vv = 0; vv < 8; ++vv) {
      acc0[vv] *= scl[vv]; acc1[vv] *= scl[vv];
      acc2[vv] *= scl[vv]; acc3[vv] *= scl[vv];
    }
    // Same-wave LDS ops are in-order: Ptile reads below see the stores above.

    // O += P @ V_tile : 4 independent WMMA chains over dh = 64
    #pragma unroll
    for (int ko = 0; ko < 16; ko += 4) {
      v2f pa = *(const v2f*)(&Ptile[w][n16 * 18 + ko + 2*hi]);  // ds_load_b64
      const float* vr0 = Vl + (ko + 2*hi) * 64;
      const float* vr1 = vr0 + 64;
      v2f b0v, b1v, b2v, b3v;
      b0v.x = vr0[ 0 + n16]; b0v.y = vr1[ 0 + n16];
      b1v.x = vr0[16 + n16]; b1v.y = vr1[16 + n16];
      b2v.x = vr0[32 + n16]; b2v.y = vr1[32 + n16];
      b3v.x = vr0[48 + n16]; b3v.y = vr1[48 + n16];
      acc0 = wmma4(pa, b0v, acc0);
      acc1 = wmma4(pa, b1v, acc1);
      acc2 = wmma4(pa, b2v, acc2);
      acc3 = wmma4(pa, b3v, acc3);
    }
  }

  // Publish this wave's partial (U, m, l) to LDS
  #pragma unroll
  for (int vv = 0; vv < 8; ++vv) {
    const int row = vv + 8*hi;
    Ubuf[w][row * 64 +  0 + n16] = acc0[vv];
    Ubuf[w][row * 64 + 16 + n16] = acc1[vv];
    Ubuf[w][row * 64 + 32 + n16] = acc2[vv];
    Ubuf[w][row * 64 + 48 + n16] = acc3[vv];
    if (n16 == 0) { Mbuf[w][row] = mrow[vv]; Lbuf[w][row] = lrow[vv]; }
  }
  __syncthreads();   // multi-wave WG: real s_barrier_signal / s_barrier_wait

  // Combine the 4 split-K partials; wave w owns output columns [w*16, w*16+16)
  const int col = w * 16 + n16;
  #pragma unroll
  for (int vv = 0; vv < 8; ++vv) {
    const int row = vv + 8*hi;
    float m0 = Mbuf[0][row], m1 = Mbuf[1][row];
    float m2 = Mbuf[2][row], m3 = Mbuf[3][row];
    float mg = fmaxf(fmaxf(m0, m1), fmaxf(m2, m3));
    float e0 = __expf(m0 - mg), e1 = __expf(m1 - mg);
    float e2 = __expf(m2 - mg), e3 = __expf(m3 - mg);
    float L  = Lbuf[0][row] * e0 + Lbuf[1][row] * e1
             + Lbuf[2][row] * e2 + Lbuf[3][row] * e3;
    float o  = (Ubuf[0][row * 64 + col] * e0 + Ubuf[1][row * 64 + col] * e1
              + Ubuf[2][row * 64 + col] * e2 + Ubuf[3][row * 64 + col] * e3)
             / L;
    Obuf[((size_t)b0 * SS_ + qt * 16 + row) * HD + h * DH + col] = o;
  }
}

// ---------------------------------------------------------------------------
// Kernel 4: out = Obuf @ W_out + b_out   (M=3072, K=512, N=512)
// ---------------------------------------------------------------------------
__global__ __launch_bounds__(32) void out_gemm_kernel(
    const float* __restrict__ A, const float* __restrict__ W,
    const float* __restrict__ bias, float* __restrict__ C)
{
  const int lane = threadIdx.x;
  const int n16  = lane & 15;
  const int hi   = lane >> 4;
  const int tm   = blockIdx.x;
  const int tn   = blockIdx.y;
  const float* arow = A + (size_t)(tm * 16 + n16) * HD;
  const int c = tn * 16 + n16;
  const float* bcol = W + c;

  v8f acc = {0,0,0,0,0,0,0,0};
  #pragma unroll 8
  for (int kk = 0; kk < HD; kk += 4) {
    v2f a = *(const v2f*)(arow + kk + 2*hi);
    v2f b;
    b.x = bcol[(size_t)(kk + 2*hi)     * DIMM];
    b.y = bcol[(size_t)(kk + 2*hi + 1) * DIMM];
    acc = wmma4(a, b, acc);
  }
  const float bo = bias[c];
  #pragma unroll
  for (int vv = 0; vv < 8; ++vv) {
    C[(size_t)(tm * 16 + vv + 8*hi) * DIMM + c] = acc[vv] + bo;
  }
}

// ---------------------------------------------------------------------------
// Host-side launcher (graph-capture safe: only kernel launches on `stream`)
// ---------------------------------------------------------------------------
extern "C" void kernel_launch(void* const* d_in, const int* in_sizes, int n_in,
                              void* d_out, int out_size, void* d_ws, size_t ws_size,
                              hipStream_t stream) {
  const float* q     = (const float*)d_in[0];
  const float* k     = (const float*)d_in[1];
  const float* v     = (const float*)d_in[2];
  const float* W_qkv = (const float*)d_in[3];
  const float* W_out = (const float*)d_in[4];
  const float* b_out = (const float*)d_in[5];
  const float* cw    = (const float*)d_in[6];
  const float* vw    = (const float*)d_in[7];
  float* out = (float*)d_out;

  // Workspace layout (~25.8 MB total, fp32) — everything L2-resident
  float* ws = (float*)d_ws;
  const size_t NR = (size_t)HEADS * BB_ * SS_;   // 24576 rows per tensor
  float* Qh    = ws; ws += NR * DH;
  float* Kh    = ws; ws += NR * DH;
  float* Vh    = ws; ws += NR * DH;
  float* Obuf  = ws; ws += (size_t)M_ROWS * HD;
  float* qmean = ws; ws += NR;
  float* qvar  = ws; ws += NR;
  float* qnorm = ws; ws += NR;
  float* kmean = ws; ws += NR;
  float* kvar  = ws; ws += NR;
  float* knorm = ws; ws += NR;

  qkv_gemm_kernel<<<dim3(M_ROWS / 16, QKV_N / 16), 32, 0, stream>>>(
      q, k, v, W_qkv, Qh, Kh, Vh);
  stats_kernel<<<(2 * (int)NR + 255) / 256, 256, 0, stream>>>(
      Qh, Kh, qmean, qvar, qnorm, kmean, kvar, knorm);
  flash_attn_kernel<<<dim3(KTILES, HEADS * BB_), NWAVE * 32, 0, stream>>>(
      Qh, Kh, Vh, qmean, qvar, qnorm, kmean, kvar, knorm, cw, vw, Obuf);
  out_gemm_kernel<<<dim3(M_ROWS / 16, DIMM / 16), 32, 0, stream>>>(
      Obuf, W_out, b_out, out);
}